// EdgeCorrGNN_66022237274283
// MI455X (gfx1250) — compile-verified
//
#include <hip/hip_runtime.h>
#include <hip/hip_bf16.h>

typedef __attribute__((ext_vector_type(2))) float v2f;
typedef __attribute__((ext_vector_type(4))) float v4f;
typedef __attribute__((ext_vector_type(8))) float v8f;

#define NNODES_F1 128
#define MAX_K 256

#define LO(v) __builtin_shufflevector((v), (v), 0, 1)
#define HI(v) __builtin_shufflevector((v), (v), 2, 3)

// ---------------------------------------------------------------------------
// WMMA fp32 GEMM: C[M,N] = A[M,K] * B[K,N]   (row-major, K%8==0, N%16==0)
// B tile (K x 16) staged in LDS transposed. Per 8-K block each lane loads one
// float4 of A per M-tile and one float4 of B (LDS); a K-permutation makes the
// .xy / .zw halves line up for two WMMA ops. Software-pipelined by one block.
// Each wave computes a 64x16 strip of C; 8 waves/block -> 512 rows x 16 cols.
// ---------------------------------------------------------------------------
__global__ __launch_bounds__(256)
void gemm_f32_wmma_kernel(const float* __restrict__ A,
                          const float* __restrict__ B,
                          float* __restrict__ C,
                          int M, int K, int N) {
    __shared__ float Bs[MAX_K * 16];     // transposed: Bs[n*K + k]

    const int tid   = threadIdx.x;
    const int lane  = tid & 31;
    const int wave  = tid >> 5;
    const int half  = lane >> 4;         // 0: lanes 0-15, 1: lanes 16-31
    const int l16   = lane & 15;
    const int tileN = blockIdx.x * 16;

    // Cooperative load of B tile into LDS (transposed), all waves participate.
    for (int i = tid; i < K * 16; i += 256) {
        const int kb = i >> 4;
        const int nb = i & 15;
        Bs[nb * K + kb] = B[(size_t)kb * N + tileN + nb];
    }
    __syncthreads();

    const int rowBase = (blockIdx.y * 8 + wave) * 64;
    if (rowBase >= M) return;            // wave-uniform: EXEC stays all-1s

    const int n = tileN + l16;
    const float* Brow = Bs + l16 * K;

    int r0 = rowBase + 0 * 16 + l16; r0 = r0 < M ? r0 : M - 1;
    int r1 = rowBase + 1 * 16 + l16; r1 = r1 < M ? r1 : M - 1;
    int r2 = rowBase + 2 * 16 + l16; r2 = r2 < M ? r2 : M - 1;
    int r3 = rowBase + 3 * 16 + l16; r3 = r3 < M ? r3 : M - 1;
    const float* A0 = A + (size_t)r0 * K;
    const float* A1 = A + (size_t)r1 * K;
    const float* A2 = A + (size_t)r2 * K;
    const float* A3 = A + (size_t)r3 * K;

    v8f acc0 = {}, acc1 = {}, acc2 = {}, acc3 = {};

    // ---- prologue: load first 8-K block ----
    const int h4 = half * 4;
    v4f ca0 = *(const v4f*)(A0 + h4);
    v4f ca1 = *(const v4f*)(A1 + h4);
    v4f ca2 = *(const v4f*)(A2 + h4);
    v4f ca3 = *(const v4f*)(A3 + h4);
    v4f cb  = *(const v4f*)(Brow + h4);

    // ---- pipelined main loop ----
    for (int k = 0; k + 8 < K; k += 8) {
        const int nk = k + 8 + h4;
        // next block's loads first: independent of current WMMAs
        v4f na0 = *(const v4f*)(A0 + nk);
        v4f na1 = *(const v4f*)(A1 + nk);
        v4f na2 = *(const v4f*)(A2 + nk);
        v4f na3 = *(const v4f*)(A3 + nk);
        v4f nb  = *(const v4f*)(Brow + nk);

        __builtin_prefetch(A0 + nk + 64, 0, 3);   // global_prefetch_b8
        __builtin_prefetch(A2 + nk + 64, 0, 3);

        acc0 = __builtin_amdgcn_wmma_f32_16x16x4_f32(false, LO(ca0), false, LO(cb),
                                                     (short)0, acc0, false, false);
        acc1 = __builtin_amdgcn_wmma_f32_16x16x4_f32(false, LO(ca1), false, LO(cb),
                                                     (short)0, acc1, false, false);
        acc2 = __builtin_amdgcn_wmma_f32_16x16x4_f32(false, LO(ca2), false, LO(cb),
                                                     (short)0, acc2, false, false);
        acc3 = __builtin_amdgcn_wmma_f32_16x16x4_f32(false, LO(ca3), false, LO(cb),
                                                     (short)0, acc3, false, false);
        acc0 = __builtin_amdgcn_wmma_f32_16x16x4_f32(false, HI(ca0), false, HI(cb),
                                                     (short)0, acc0, false, false);
        acc1 = __builtin_amdgcn_wmma_f32_16x16x4_f32(false, HI(ca1), false, HI(cb),
                                                     (short)0, acc1, false, false);
        acc2 = __builtin_amdgcn_wmma_f32_16x16x4_f32(false, HI(ca2), false, HI(cb),
                                                     (short)0, acc2, false, false);
        acc3 = __builtin_amdgcn_wmma_f32_16x16x4_f32(false, HI(ca3), false, HI(cb),
                                                     (short)0, acc3, false, false);

        ca0 = na0; ca1 = na1; ca2 = na2; ca3 = na3; cb = nb;
    }

    // ---- epilogue: last 8-K block ----
    acc0 = __builtin_amdgcn_wmma_f32_16x16x4_f32(false, LO(ca0), false, LO(cb),
                                                 (short)0, acc0, false, false);
    acc1 = __builtin_amdgcn_wmma_f32_16x16x4_f32(false, LO(ca1), false, LO(cb),
                                                 (short)0, acc1, false, false);
    acc2 = __builtin_amdgcn_wmma_f32_16x16x4_f32(false, LO(ca2), false, LO(cb),
                                                 (short)0, acc2, false, false);
    acc3 = __builtin_amdgcn_wmma_f32_16x16x4_f32(false, LO(ca3), false, LO(cb),
                                                 (short)0, acc3, false, false);
    acc0 = __builtin_amdgcn_wmma_f32_16x16x4_f32(false, HI(ca0), false, HI(cb),
                                                 (short)0, acc0, false, false);
    acc1 = __builtin_amdgcn_wmma_f32_16x16x4_f32(false, HI(ca1), false, HI(cb),
                                                 (short)0, acc1, false, false);
    acc2 = __builtin_amdgcn_wmma_f32_16x16x4_f32(false, HI(ca2), false, HI(cb),
                                                 (short)0, acc2, false, false);
    acc3 = __builtin_amdgcn_wmma_f32_16x16x4_f32(false, HI(ca3), false, HI(cb),
                                                 (short)0, acc3, false, false);

    // Store: C/D 16x16 layout -> VGPR i holds row (i + half*8), col = lane%16
    v8f accs[4] = {acc0, acc1, acc2, acc3};
    #pragma unroll
    for (int t = 0; t < 4; ++t) {
        #pragma unroll
        for (int i = 0; i < 8; ++i) {
            const int row = rowBase + t * 16 + i + half * 8;
            if (row < M) C[(size_t)row * N + n] = accs[t][i];
        }
    }
}

// ---------------------------------------------------------------------------
// Degree / normalization
// ---------------------------------------------------------------------------
__global__ void deg_init_kernel(float* __restrict__ deg, int n) {
    int i = blockIdx.x * blockDim.x + threadIdx.x;
    if (i < n) deg[i] = 1.0f;  // self-loop weight
}

__global__ void deg_edges_kernel(const long long* __restrict__ dst,
                                 const float* __restrict__ w,
                                 float* __restrict__ deg, int E) {
    int e = blockIdx.x * blockDim.x + threadIdx.x;
    if (e < E) atomicAdd(&deg[(int)dst[e]], w[e]);
}

__global__ void deg_rsqrt_kernel(float* __restrict__ deg, int n) {
    int i = blockIdx.x * blockDim.x + threadIdx.x;
    if (i < n) {
        float d = deg[i];
        deg[i] = d > 0.0f ? rsqrtf(d) : 0.0f;   // becomes dinv in-place
    }
}

// Precompute per-edge int32 indices and normalization coefficient
// coef[e] = dinv[src]*w[e]*dinv[dst]  (reused by all 4 layers)
__global__ void edge_prep_kernel(const long long* __restrict__ src,
                                 const long long* __restrict__ dst,
                                 const float* __restrict__ w,
                                 const float* __restrict__ dinv,
                                 int* __restrict__ src32,
                                 int* __restrict__ dst32,
                                 float* __restrict__ coef, int E) {
    int e = blockIdx.x * blockDim.x + threadIdx.x;
    if (e < E) {
        int s = (int)src[e];
        int d = (int)dst[e];
        src32[e] = s;
        dst32[e] = d;
        coef[e] = dinv[s] * w[e] * dinv[d];
    }
}

// ---------------------------------------------------------------------------
// Aggregation: out = scatter_add(h[src]*coef) with self loops, then bias+ReLU
// ---------------------------------------------------------------------------
__global__ void agg_self_kernel(const float* __restrict__ hw,
                                const float* __restrict__ dinv,
                                float* __restrict__ out, int n, int F) {
    int idx = blockIdx.x * blockDim.x + threadIdx.x;
    if (idx < n * F) {
        int i = idx / F;
        float di = dinv[i];
        out[idx] = hw[idx] * di * di;    // self-loop term (w=1)
    }
}

__global__ void agg_edges_kernel(const int* __restrict__ src32,
                                 const int* __restrict__ dst32,
                                 const float* __restrict__ coef,
                                 const float* __restrict__ hw,
                                 float* __restrict__ out,
                                 int E, int F, int lc /* log2(F/4) */) {
    int tid = blockIdx.x * blockDim.x + threadIdx.x;
    int e = tid >> lc;
    if (e >= E) return;
    int c = (tid & ((1 << lc) - 1)) << 2;
    int s = src32[e];
    int d = dst32[e];
    float cf = coef[e];
    const float4 hv = *(const float4*)(hw + (size_t)s * F + c);
    float* o = out + (size_t)d * F + c;
    atomicAdd(o + 0, hv.x * cf);
    atomicAdd(o + 1, hv.y * cf);
    atomicAdd(o + 2, hv.z * cf);
    atomicAdd(o + 3, hv.w * cf);
}

__global__ void bias_relu_kernel(float* __restrict__ out,
                                 const float* __restrict__ b,
                                 int total, int fmask) {
    int idx = blockIdx.x * blockDim.x + threadIdx.x;
    if (idx < total) {
        float v = out[idx] + b[idx & fmask];
        out[idx] = v > 0.0f ? v : 0.0f;
    }
}

// ---------------------------------------------------------------------------
// Final projection: out[i] = dot(h[i,:256], Wf) + bf   (one wave per node)
// ---------------------------------------------------------------------------
__global__ __launch_bounds__(256)
void final_dot_kernel(const float* __restrict__ h,
                      const float* __restrict__ Wf,
                      const float* __restrict__ bf,
                      float* __restrict__ out, int n) {
    int node = blockIdx.x * (blockDim.x >> 5) + (threadIdx.x >> 5);
    int lane = threadIdx.x & 31;
    if (node >= n) return;
    const float* row = h + (size_t)node * 256;
    float s = 0.0f;
    #pragma unroll
    for (int f = 0; f < 256; f += 32) s += row[f + lane] * Wf[f + lane];
    #pragma unroll
    for (int off = 16; off > 0; off >>= 1) s += __shfl_xor(s, off, 32);
    if (lane == 0) out[node] = s + bf[0];
}

// ---------------------------------------------------------------------------
// Host orchestration
// ---------------------------------------------------------------------------
static inline int ceil_div(int a, int b) { return (a + b - 1) / b; }

extern "C" void kernel_launch(void* const* d_in, const int* in_sizes, int n_in,
                              void* d_out, int out_size, void* d_ws, size_t ws_size,
                              hipStream_t stream) {
    const float*     x    = (const float*)d_in[0];
    const long long* eidx = (const long long*)d_in[1];
    const float*     ew   = (const float*)d_in[2];
    const float*     Wl[4] = {(const float*)d_in[3], (const float*)d_in[5],
                              (const float*)d_in[7], (const float*)d_in[9]};
    const float*     bl[4] = {(const float*)d_in[4], (const float*)d_in[6],
                              (const float*)d_in[8], (const float*)d_in[10]};
    const float*     Wf   = (const float*)d_in[11];
    const float*     bf   = (const float*)d_in[12];
    float*           out  = (float*)d_out;

    const int N = in_sizes[0] / NNODES_F1;   // 50000
    const int E = in_sizes[2];               // 800000
    const long long* esrc = eidx;
    const long long* edst = eidx + E;

    // Workspace: dinv[N] | coef[E] | src32[E] | dst32[E] | B0[N*256] | B1[N*256]
    char* ws = (char*)d_ws;
    size_t off = 0;
    auto take = [&](size_t bytes) -> char* {
        char* p = ws + off;
        off = (off + bytes + 255) & ~(size_t)255;
        return p;
    };
    float* dinv  = (float*)take((size_t)N * sizeof(float));
    float* coef  = (float*)take((size_t)E * sizeof(float));
    int*   src32 = (int*)take((size_t)E * sizeof(int));
    int*   dst32 = (int*)take((size_t)E * sizeof(int));
    float* B0    = (float*)take((size_t)N * 256 * sizeof(float));
    float* B1    = (float*)take((size_t)N * 256 * sizeof(float));

    const int T = 256;

    // ---- degree -> dinv -> per-edge coefficients ----
    deg_init_kernel<<<ceil_div(N, T), T, 0, stream>>>(dinv, N);
    deg_edges_kernel<<<ceil_div(E, T), T, 0, stream>>>(edst, ew, dinv, E);
    deg_rsqrt_kernel<<<ceil_div(N, T), T, 0, stream>>>(dinv, N);
    edge_prep_kernel<<<ceil_div(E, T), T, 0, stream>>>(esrc, edst, ew, dinv,
                                                       src32, dst32, coef, E);

    // ---- 4 GCN layers ----
    const int fin[4]  = {128, 64, 128, 256};
    const int fout[4] = {64, 128, 256, 256};
    const float* hin = x;
    for (int l = 0; l < 4; ++l) {
        const int K = fin[l], F = fout[l];

        // GEMM: B0 = hin @ Wl
        dim3 ggrid(F / 16, ceil_div(N, 512));
        gemm_f32_wmma_kernel<<<ggrid, T, 0, stream>>>(hin, Wl[l], B0, N, K, F);

        // Aggregation into B1
        agg_self_kernel<<<ceil_div(N * F, T), T, 0, stream>>>(B0, dinv, B1, N, F);
        int lc = (F == 64) ? 4 : (F == 128) ? 5 : 6;  // log2(F/4)
        int work = E << lc;
        agg_edges_kernel<<<ceil_div(work, T), T, 0, stream>>>(
            src32, dst32, coef, B0, B1, E, F, lc);
        bias_relu_kernel<<<ceil_div(N * F, T), T, 0, stream>>>(B1, bl[l], N * F, F - 1);

        hin = B1;
    }

    // ---- final 256 -> 1 projection ----
    final_dot_kernel<<<ceil_div(N, T / 32), T, 0, stream>>>(B1, Wf, bf, out, N);
}